// CapsuleConv3D_36215164240456
// MI455X (gfx1250) — compile-verified
//
#include <hip/hip_runtime.h>
#include <hip/hip_bf16.h>

// CapsuleConv3D EM-routing, fully fused per output voxel.
// - 1000 workgroups (one per output voxel), 256 threads (8 wave32s).
// - Votes computed with V_WMMA_F32_16X16X4_F32 (fp32 exact) into LDS.
// - 3 EM iterations run entirely out of LDS (~260 KB of the 320 KB/WGP).

typedef __attribute__((ext_vector_type(2))) float v2f;
typedef __attribute__((ext_vector_type(8))) float v8f;

#define NI 216            // 27 kernel offsets * 8 input capsules
#define NO 16             // output capsules
#define RC 16             // 4x4 pose entries
#define OD 10             // output spatial dim (12 - 3 + 1)
#define EPS 1e-9f
#define LOG2PI 1.8378770664093453f
#define NEG_LOG16 -2.7725887222397811f

// LDS float counts
#define SM_VOTES   (NI * 256)     // votes[i][o*16+rc]
#define SM_PB      (NI * 16)      // pose block  pb[i][k*4+c]
#define SM_LGT     (NI * NO)      // nR weights / e-step logits (reused)
#define SM_LOGR    (NI * NO)
#define SMEM_FLOATS (SM_VOTES + SM_PB + SM_LGT + SM_LOGR + 256 + 256 + 256 + NI + 16 + 16)

__global__ __launch_bounds__(256)
void caps_em_kernel(const float* __restrict__ g_pose,   // (1,12,12,12,8,4,4)
                    const float* __restrict__ g_act,    // (1,12,12,12,8)
                    const float* __restrict__ g_W,      // (216,16,4,4)
                    const float* __restrict__ g_bv,     // (16,1)
                    const float* __restrict__ g_ba,     // (16,1)
                    float* __restrict__ out_pose,       // (1000,16,4,4)
                    float* __restrict__ out_act)        // (1000,16)
{
    extern __shared__ float smem[];
    float* votes = smem;                   // [NI][256]
    float* pb    = votes + SM_VOTES;       // [NI][16]
    float* lgt   = pb    + SM_PB;          // [NI][16]  normalized R (m-step) / logits (e-step)
    float* logR  = lgt   + SM_LGT;         // [NI][16]
    float* poseL = logR  + SM_LOGR;        // [256]  pose[o*16+rc]
    float* lvarL = poseL + 256;            // [256]  log_var
    float* ivarL = lvarL + 256;            // [256]  1/var
    float* lactL = ivarL + 256;            // [NI]   log(act+eps)
    float* RsumL = lactL + NI;             // [16]
    float* logaL = RsumL + 16;             // [16]

    const int tid = threadIdx.x;
    const int pos = blockIdx.x;
    const int ow = pos % OD, oh = (pos / OD) % OD, od = pos / (OD * OD);

    // ---- Stage pose receptive field + log-activations into LDS ----
    for (int idx = tid; idx < NI * 16; idx += 256) {
        int i = idx >> 4, kc = idx & 15;
        int kd = i / 72, kh = (i / 24) % 3, kw = (i / 8) % 3, ci = i & 7;
        size_t g = ((((size_t)(od + kd) * 12 + (oh + kh)) * 12 + (ow + kw)) * 8 + ci) * 16 + kc;
        pb[idx] = g_pose[g];
    }
    for (int i = tid; i < NI; i += 256) {
        int kd = i / 72, kh = (i / 24) % 3, kw = (i / 8) % 3, ci = i & 7;
        size_t g = (((size_t)(od + kd) * 12 + (oh + kh)) * 12 + (ow + kw)) * 8 + ci;
        lactL[i] = __logf(g_act[g] + EPS);
    }
    // init log_R = -log(16)
    for (int idx = tid; idx < NI * NO; idx += 256) logR[idx] = NEG_LOG16;
    __syncthreads();

    // ---- Votes: vote[i][o*16 + r*4 + c] = sum_k W[i,o,r,k] * pb[i,k,c] ----
    // One V_WMMA_F32_16X16X4_F32 per (i, r):
    //   A[o,k] = W[i,o,r,k]   (16x4)   lanes 0-15: K=0,1 ; lanes 16-31: K=2,3 ; M = lane&15
    //   B[k,c] = pb[i,k,c]    (4x16, cols c>=4 zero-padded)
    //   D[o,c] 16x16 f32: VGPR j -> M = j + 8*(lane>>4), N = lane&15
    {
        const int lane = tid & 31;
        const int wave = tid >> 5;
        const int mrow = lane & 15;          // A: o row / B,D: column c
        const int k0   = (lane >> 4) << 1;   // K start: 0 or 2
        for (int p = wave; p < NI * 4; p += 8) {
            int i = p >> 2, r = p & 3;
            v2f a = *(const v2f*)(g_W + ((size_t)(i * 16 + mrow) * 16 + r * 4 + k0));
            v2f b;
            if (mrow < 4) {
                b.x = pb[i * 16 + k0 * 4 + mrow];
                b.y = pb[i * 16 + (k0 + 1) * 4 + mrow];
            } else {
                b.x = 0.0f; b.y = 0.0f;
            }
            v8f c = {};
            v8f d = __builtin_amdgcn_wmma_f32_16x16x4_f32(
                false, a, false, b, (short)0, c, false, false);
            if (mrow < 4) {
                int base  = i * 256 + r * 4 + mrow;   // + o*16
                int ohalf = (lane >> 4) * 8;
#pragma unroll
                for (int j = 0; j < 8; ++j)
                    votes[base + (j + ohalf) * 16] = d[j];
            }
        }
    }
    __syncthreads();

    // ---- EM routing: 3 iterations (m, e, m, e, m) ----
    for (int it = 0; it < 3; ++it) {
        // m-step. w[i,o] = exp(log_R + log_act)
        for (int idx = tid; idx < NI * NO; idx += 256)
            lgt[idx] = __expf(logR[idx] + lactL[idx >> 4]);
        __syncthreads();
        if (tid < NO) {                       // R_sum[o] = sum_i w
            float s = 0.0f;
            for (int i = 0; i < NI; ++i) s += lgt[i * NO + tid];
            RsumL[tid] = s;
        }
        __syncthreads();
        for (int idx = tid; idx < NI * NO; idx += 256)   // normalize: nR
            lgt[idx] = lgt[idx] / RsumL[idx & 15];
        __syncthreads();
        {   // pose[o,rc] = sum_i nR[i,o] * vote[i,o,rc]
            int o = tid >> 4;
            float s = 0.0f;
            for (int i = 0; i < NI; ++i)
                s += votes[i * 256 + tid] * lgt[i * NO + o];
            poseL[tid] = s;
        }
        __syncthreads();
        {   // var[o,rc] = sum_i nR * ((v-p)^2 + eps);  log_var = log(var)
            int o = tid >> 4;
            float pz = poseL[tid];
            float s = 0.0f;
            for (int i = 0; i < NI; ++i) {
                float dv = votes[i * 256 + tid] - pz;
                s += lgt[i * NO + o] * (dv * dv + EPS);
            }
            lvarL[tid] = __logf(s);
            ivarL[tid] = 1.0f / s;
        }
        __syncthreads();
        if (tid < NO) {                       // cost -> logit -> log_sigmoid
            float bv = g_bv[tid], ba = g_ba[tid], rs = RsumL[tid];
            float cs = 0.0f;
            for (int rc = 0; rc < RC; ++rc)
                cs += rs * (bv + 0.5f * lvarL[tid * 16 + rc]);
            float lg = 0.01f * (ba - cs);
            logaL[tid] = fminf(lg, 0.0f) - log1pf(__expf(-fabsf(lg)));
        }
        __syncthreads();
        if (it == 2) break;

        // e-step. logit[i,o] = log_a[o] - 0.5*sum_rc((v-p)^2/var + log_var + log(2pi))
        for (int p = tid; p < NI * NO; p += 256) {
            int i = p >> 4, o = p & 15;
            float lp = 0.0f;
            for (int rc = 0; rc < RC; ++rc) {
                float dv = votes[i * 256 + o * 16 + rc] - poseL[o * 16 + rc];
                lp += dv * dv * ivarL[o * 16 + rc] + lvarL[o * 16 + rc];
            }
            lgt[p] = logaL[o] - 0.5f * (lp + 16.0f * LOG2PI);
        }
        __syncthreads();
        for (int i = tid; i < NI; i += 256) { // log_R = logit - lse_o(logit)
            float m = -3.0e38f;
            for (int o = 0; o < NO; ++o) m = fmaxf(m, lgt[i * 16 + o]);
            float s = 0.0f;
            for (int o = 0; o < NO; ++o) s += __expf(lgt[i * 16 + o] - m);
            float l = m + __logf(s);
            for (int o = 0; o < NO; ++o) logR[i * 16 + o] = lgt[i * 16 + o] - l;
        }
        __syncthreads();
    }

    // ---- Emit outputs (each thread wrote its own poseL[tid]/logaL[tid]) ----
    out_pose[(size_t)pos * 256 + tid] = poseL[tid];
    if (tid < NO)
        out_act[(size_t)pos * 16 + tid] = __expf(logaL[tid]);
}

extern "C" void kernel_launch(void* const* d_in, const int* in_sizes, int n_in,
                              void* d_out, int out_size, void* d_ws, size_t ws_size,
                              hipStream_t stream) {
    const float* g_pose = (const float*)d_in[0];
    const float* g_act  = (const float*)d_in[1];
    const float* g_W    = (const float*)d_in[2];
    const float* g_bv   = (const float*)d_in[3];
    const float* g_ba   = (const float*)d_in[4];
    float* op = (float*)d_out;              // 1000*16*16 pose floats
    float* oa = op + 1000 * 256;            // then 1000*16 activations

    const size_t shmem = (size_t)SMEM_FLOATS * sizeof(float);  // ~260 KB (<320 KB/WGP)
    (void)hipFuncSetAttribute((const void*)caps_em_kernel,
                              hipFuncAttributeMaxDynamicSharedMemorySize,
                              (int)shmem);
    caps_em_kernel<<<1000, 256, shmem, stream>>>(g_pose, g_act, g_W, g_bv, g_ba, op, oa);
}